// MultiBox_10522669875218
// MI455X (gfx1250) — compile-verified
//
#include <hip/hip_runtime.h>
#include <math.h>
#include <stdint.h>

// ---------------- problem constants (match reference) ----------------
#define BATCH   8
#define NANCH   120000
#define NCLS    21
#define NSCORE  20
#define MCAND   4096
#define TOPN    100
#define SCORE_T 0.5f
#define NMS_T   0.45f
#define VAR0f   0.1f
#define VAR1f   0.2f

typedef unsigned int u32;
typedef unsigned long long u64;
typedef u32 u32x4 __attribute__((ext_vector_type(4)));
typedef int  i32x4 __attribute__((ext_vector_type(4)));
typedef int  i32x8 __attribute__((ext_vector_type(8)));

#if defined(__gfx1250__) && __has_builtin(__builtin_amdgcn_tensor_load_to_lds)
#define HAVE_TDM 1
#else
#define HAVE_TDM 0
#endif

// ---------------- box decode (shared by select/filler passes) --------
__device__ __forceinline__ void decode_box(const float* __restrict__ deltas,
                                           const float* __restrict__ priors,
                                           int b, int n, float box[4]) {
  const float* d = deltas + ((size_t)b * NANCH + (size_t)n) * 4;
  const float* p = priors + (size_t)n * 4;
  float pcx = p[0], pcy = p[1], pw = p[2], ph = p[3];
  float cx = d[0] * VAR0f * pw + pcx;
  float cy = d[1] * VAR0f * ph + pcy;
  float w  = expf(d[2] * VAR1f) * pw;
  float h  = expf(d[3] * VAR1f) * ph;
  box[0] = cx - 0.5f * w;
  box[1] = cy - 0.5f * h;
  box[2] = cx + 0.5f * w;
  box[3] = cy + 0.5f * h;
}

// ---------------- TDM: 1-D f32 tile, global -> LDS (ISA 08, D# §8.3/8.4)
#if HAVE_TDM
__device__ __forceinline__ void tdm_copy_f32(const void* gptr, u32 lds_byte_off, u32 nelem) {
  u64 ga = (u64)(uintptr_t)gptr;
  // Group0: count=1 | lds_addr | global_addr[31:0] | {type=2, global_addr[56:32]}
  u32x4 g0 = { 1u,
               lds_byte_off,
               (u32)(ga & 0xFFFFFFFFull),
               (((u32)(ga >> 32)) & 0x01FFFFFFu) | (2u << 30) };
  // Group1: data_size=4B (bits17:16); tensor_dim0=nelem (bits79:48);
  // tensor_dim1=1 (bits111:80); tile_dim0=nelem (bits127:112); tile_dim1=1;
  // tensor_dim0_stride=nelem (bits207:160); rest zero.
  i32x8 g1 = { (int)(2u << 16),
               (int)((nelem & 0xFFFFu) << 16),
               (int)((((nelem >> 16) & 0xFFFFu)) | (1u << 16)),
               (int)((nelem & 0xFFFFu) << 16),
               (int)1,
               (int)nelem,
               0, 0 };
  i32x4 z4 = { 0, 0, 0, 0 };
#if defined(__clang_major__) && (__clang_major__ >= 23)
  i32x8 z8 = { 0, 0, 0, 0, 0, 0, 0, 0 };
  __builtin_amdgcn_tensor_load_to_lds(g0, g1, z4, z4, z8, 0);
#else
  __builtin_amdgcn_tensor_load_to_lds(g0, g1, z4, z4, 0);
#endif
}
#endif

// ---------------- Kernel 1: streaming softmax + best-class filter ----
// Bandwidth-bound (~88MB at 23.3TB/s). At most one class can have
// softmax prob > 0.5, so the per-anchor max class is exact.
__global__ __launch_bounds__(256) void k_softmax_filter(
    const float* __restrict__ obj,
    float* __restrict__ scoreBuf, int* __restrict__ labelBuf) {
  int idx = blockIdx.x * 256 + threadIdx.x;
  if (idx >= BATCH * NANCH) return;
  const float* row = obj + (size_t)idx * NCLS;
  __builtin_prefetch(row + (size_t)256 * NCLS, 0, 1);  // global_prefetch_b8
  float m = row[0];
#pragma unroll
  for (int c = 1; c < NCLS; ++c) m = fmaxf(m, row[c]);
  float sum = 0.0f;
  float bv = row[1]; int bc = 1;
#pragma unroll
  for (int c = 0; c < NCLS; ++c) {
    float l = row[c];
    sum += expf(l - m);
    if (c >= 1 && l > bv) { bv = l; bc = c; }
  }
  float prob = expf(bv - m) / sum;
  scoreBuf[idx] = (prob > SCORE_T) ? prob : -1.0f;
  labelBuf[idx] = bc;  // label in 1..20, matches (flat % 20) + 1
}

// ---------------- Kernel 2 helpers: exact top-M via bit bisection ----
__device__ int block_count_gt(const float* __restrict__ sb, int b, u32 ubits, int* shAcc) {
  if (threadIdx.x == 0) *shAcc = 0;
  __syncthreads();
  int local = 0;
  for (int n = threadIdx.x; n < NANCH; n += 1024) {
    float s = sb[(size_t)b * NANCH + n];
    if (s > SCORE_T && __float_as_uint(s) > ubits) ++local;
  }
  atomicAdd(shAcc, local);
  __syncthreads();
  int r = *shAcc;
  __syncthreads();
  return r;
}

// Index-stable compaction via wave32 ballot prefix sums.
// pass 0: bits >  vbits   (strictly above cutoff)
// pass 1: bits == vbits   (ties at cutoff, lowest index first)
// pass 2: filler (-1.0 entries), flat-index order, matching jax top_k padding
__device__ void compact_pass(int pass, int b, u32 vbits, int baseSlot, int cap,
                             const float* __restrict__ scoreBuf,
                             const int* __restrict__ labelBuf,
                             const float* __restrict__ deltas,
                             const float* __restrict__ priors,
                             float* __restrict__ candScore,
                             int* __restrict__ candLabel,
                             float* __restrict__ candBox,
                             int* shWaveCnt /* [33] */) {
  if (cap <= 0) return;
  const int tid = threadIdx.x;
  const int lane = tid & 31;
  const int wv = tid >> 5;
  const long domain = (pass == 2) ? (long)NANCH * NSCORE : (long)NANCH;
  if (tid == 0) shWaveCnt[32] = 0;
  __syncthreads();
  for (long base = 0; base < domain; base += 1024) {
    int done = shWaveCnt[32];
    if (done >= cap) break;  // uniform
    long i = base + tid;
    bool pred = false;
    int n = 0, lab = 0; float sc = 0.0f;
    if (i < domain) {
      if (pass == 2) {
        n = (int)(i / NSCORE);
        int c = (int)(i % NSCORE);
        lab = c + 1;
        sc = -1.0f;
        float s = scoreBuf[(size_t)b * NANCH + n];
        int bl = labelBuf[(size_t)b * NANCH + n];
        pred = !((s > SCORE_T) && (bl == lab));  // not already a candidate
      } else {
        n = (int)i;
        float s = scoreBuf[(size_t)b * NANCH + n];
        u32 bits = __float_as_uint(s);
        bool cand = (s > SCORE_T);
        pred = cand && ((pass == 0) ? (bits > vbits) : (bits == vbits));
        lab = labelBuf[(size_t)b * NANCH + n];
        sc = s;
      }
    }
    u32 m32 = (u32)__ballot(pred);                 // wave32: low 32 bits
    int prefix = __popc(m32 & ((1u << lane) - 1u));
    if (lane == 0) shWaveCnt[wv] = __popc(m32);
    __syncthreads();
    int waveOff = 0;
    for (int w = 0; w < wv; ++w) waveOff += shWaveCnt[w];
    int rank = done + waveOff + prefix;
    if (pred && rank < cap) {
      int slot = baseSlot + rank;
      size_t o = (size_t)b * MCAND + (size_t)slot;
      candScore[o] = sc;
      candLabel[o] = lab;
      float bx[4];
      decode_box(deltas, priors, b, n, bx);
#pragma unroll
      for (int k = 0; k < 4; ++k)                  // SoA: conflict-free LDS reads later
        candBox[((size_t)b * 4 + k) * MCAND + slot] = bx[k];
    }
    __syncthreads();
    if (tid == 0) {
      int t = 0;
      for (int w = 0; w < 32; ++w) t += shWaveCnt[w];
      shWaveCnt[32] = done + t;
    }
    __syncthreads();
  }
}

// ---------------- Kernel 2: per-batch exact top-4096 selection -------
__global__ __launch_bounds__(1024) void k_select(
    const float* __restrict__ scoreBuf, const int* __restrict__ labelBuf,
    const float* __restrict__ deltas, const float* __restrict__ priors,
    float* __restrict__ candScore, int* __restrict__ candLabel,
    float* __restrict__ candBox) {
  __shared__ int shAcc;
  __shared__ int shWaveCnt[33];
  const int b = blockIdx.x;
  const u32 T_BITS = 0x3F000000u;  // bits of 0.5f; scores in (0.5,1) are bit-monotone
  int total = block_count_gt(scoreBuf, b, T_BITS, &shAcc);
  u32 v = T_BITS;
  int nGreater = total, eqNeeded = 0, filler = 0;
  if (total > MCAND) {
    // minimal u with #{bits > u} < M  ==  bits of the M-th largest score
    u32 lo = T_BITS, hi = 0x3F800000u;
    while (lo < hi) {
      u32 mid = lo + ((hi - lo) >> 1);
      int c = block_count_gt(scoreBuf, b, mid, &shAcc);
      if (c < MCAND) hi = mid; else lo = mid + 1u;
    }
    v = lo;
    nGreater = block_count_gt(scoreBuf, b, v, &shAcc);
    eqNeeded = MCAND - nGreater;
  } else {
    filler = MCAND - total;
  }
  compact_pass(0, b, v, 0, nGreater, scoreBuf, labelBuf, deltas, priors,
               candScore, candLabel, candBox, shWaveCnt);
  compact_pass(1, b, v, nGreater, eqNeeded, scoreBuf, labelBuf, deltas, priors,
               candScore, candLabel, candBox, shWaveCnt);
  compact_pass(2, b, v, total, filler, scoreBuf, labelBuf, deltas, priors,
               candScore, candLabel, candBox, shWaveCnt);
}

// ---------------- Kernel 3: LDS-resident sequential NMS --------------
// 96KB candidate set staged via TDM; 100 rounds of argmax + suppress.
#define SMEM_BOX    0
#define SMEM_SCORE  65536
#define SMEM_LABEL  81920
#define SMEM_REDV   98304
#define SMEM_REDI   (98304 + 128)
#define SMEM_BB     (98304 + 256)
#define SMEM_BL     (98304 + 272)
#define SMEM_TOTAL  98560

__global__ __launch_bounds__(1024) void k_nms(
    const float* __restrict__ candScore, const int* __restrict__ candLabel,
    const float* __restrict__ candBox, float* __restrict__ out) {
  extern __shared__ unsigned char smem[];
  float* sBox   = (float*)(smem + SMEM_BOX);    // [4][MCAND] SoA
  float* sScore = (float*)(smem + SMEM_SCORE);  // [MCAND]
  int*   sLabel = (int*)(smem + SMEM_LABEL);    // [MCAND]
  float* sRedV  = (float*)(smem + SMEM_REDV);   // 32
  int*   sRedI  = (int*)(smem + SMEM_REDI);     // 32
  float* sBB    = (float*)(smem + SMEM_BB);     // best box broadcast
  int*   sBL    = (int*)(smem + SMEM_BL);       // best label broadcast

  const int b = blockIdx.x;
  const float* gBox   = candBox + (size_t)b * 4 * MCAND;
  const float* gScore = candScore + (size_t)b * MCAND;
  const int*   gLabel = candLabel + (size_t)b * MCAND;

#if HAVE_TDM
  if (threadIdx.x < 32) {  // wave 0 issues the three DMA descriptors
    tdm_copy_f32(gBox,   SMEM_BOX,   4 * MCAND);
    tdm_copy_f32(gScore, SMEM_SCORE, MCAND);
    tdm_copy_f32(gLabel, SMEM_LABEL, MCAND);
    __builtin_amdgcn_s_wait_tensorcnt(0);
  }
#else
  for (int i = threadIdx.x; i < 4 * MCAND; i += 1024) sBox[i] = gBox[i];
  for (int i = threadIdx.x; i < MCAND; i += 1024) {
    sScore[i] = gScore[i];
    sLabel[i] = gLabel[i];
  }
#endif
  __syncthreads();

  const int lane = threadIdx.x & 31;
  const int wv = threadIdx.x >> 5;

  for (int t = 0; t < TOPN; ++t) {
    // ---- block argmax with first-occurrence tie-break (matches jnp.argmax)
    float lv = -1.0e30f; int li = 0;
    for (int j = threadIdx.x; j < MCAND; j += 1024) {
      float s = sScore[j];
      if (s > lv) { lv = s; li = j; }
    }
    for (int off = 16; off > 0; off >>= 1) {
      float ov = __shfl_down(lv, off, 32);
      int   oi = __shfl_down(li, off, 32);
      if (ov > lv || (ov == lv && oi < li)) { lv = ov; li = oi; }
    }
    if (lane == 0) { sRedV[wv] = lv; sRedI[wv] = li; }
    __syncthreads();
    if (threadIdx.x < 32) {
      lv = sRedV[lane]; li = sRedI[lane];
      for (int off = 16; off > 0; off >>= 1) {
        float ov = __shfl_down(lv, off, 32);
        int   oi = __shfl_down(li, off, 32);
        if (ov > lv || (ov == lv && oi < li)) { lv = ov; li = oi; }
      }
      if (lane == 0) {
        int bi = li; float bsc = lv;
        float bx0 = sBox[0 * MCAND + bi];
        float bx1 = sBox[1 * MCAND + bi];
        float bx2 = sBox[2 * MCAND + bi];
        float bx3 = sBox[3 * MCAND + bi];
        int blab = sLabel[bi];
        sBB[0] = bx0; sBB[1] = bx1; sBB[2] = bx2; sBB[3] = bx3;
        sBL[0] = blab;
        float vf = (bsc > SCORE_T) ? 1.0f : 0.0f;
        size_t bo = ((size_t)b * TOPN + t) * 4;
        out[bo + 0] = fminf(fmaxf(bx0, 0.0f), 1.0f) * vf;
        out[bo + 1] = fminf(fmaxf(bx1, 0.0f), 1.0f) * vf;
        out[bo + 2] = fminf(fmaxf(bx2, 0.0f), 1.0f) * vf;
        out[bo + 3] = fminf(fmaxf(bx3, 0.0f), 1.0f) * vf;
        out[BATCH * TOPN * 4 + b * TOPN + t] = bsc * vf;
        out[BATCH * TOPN * 5 + b * TOPN + t] = (vf > 0.0f) ? (float)blab : 0.0f;
        out[BATCH * TOPN * 6 + b * TOPN + t] = vf;
      }
    }
    __syncthreads();
    // ---- suppress same-label boxes with IoU > 0.45 (includes the winner)
    float b0 = sBB[0], b1 = sBB[1], b2 = sBB[2], b3 = sBB[3];
    int blab = sBL[0];
    float a1 = (b2 - b0) * (b3 - b1);
    for (int j = threadIdx.x; j < MCAND; j += 1024) {
      float c0 = sBox[0 * MCAND + j];
      float c1 = sBox[1 * MCAND + j];
      float c2 = sBox[2 * MCAND + j];
      float c3 = sBox[3 * MCAND + j];
      float ltx = fmaxf(b0, c0), lty = fmaxf(b1, c1);
      float rbx = fminf(b2, c2), rby = fminf(b3, c3);
      float iw = fmaxf(rbx - ltx, 0.0f), ih = fmaxf(rby - lty, 0.0f);
      float inter = iw * ih;
      float a2 = (c2 - c0) * (c3 - c1);
      float iou = inter / (a1 + a2 - inter);
      if (iou > NMS_T && sLabel[j] == blab) sScore[j] = -2.0f;
    }
    __syncthreads();
  }
}

// ---------------- host entry ----------------
extern "C" void kernel_launch(void* const* d_in, const int* in_sizes, int n_in,
                              void* d_out, int out_size, void* d_ws, size_t ws_size,
                              hipStream_t stream) {
  const float* obj    = (const float*)d_in[0];  // [B][N][21]
  const float* deltas = (const float*)d_in[1];  // [B][N][4]
  const float* priors = (const float*)d_in[2];  // [N][4]
  float* out = (float*)d_out;                   // 3200 box + 800 score + 800 label + 800 valid

  char* ws = (char*)d_ws;
  const size_t SB = 0;
  const size_t LB = SB + (size_t)BATCH * NANCH * sizeof(float);
  const size_t CS = LB + (size_t)BATCH * NANCH * sizeof(int);
  const size_t CL = CS + (size_t)BATCH * MCAND * sizeof(float);
  const size_t CB = CL + (size_t)BATCH * MCAND * sizeof(int);
  float* scoreBuf  = (float*)(ws + SB);
  int*   labelBuf  = (int*)(ws + LB);
  float* candScore = (float*)(ws + CS);
  int*   candLabel = (int*)(ws + CL);
  float* candBox   = (float*)(ws + CB);  // [B][4][M] SoA

  k_softmax_filter<<<dim3((BATCH * NANCH + 255) / 256), dim3(256), 0, stream>>>(
      obj, scoreBuf, labelBuf);
  k_select<<<dim3(BATCH), dim3(1024), 0, stream>>>(
      scoreBuf, labelBuf, deltas, priors, candScore, candLabel, candBox);
  k_nms<<<dim3(BATCH), dim3(1024), SMEM_TOTAL, stream>>>(
      candScore, candLabel, candBox, out);

  (void)in_sizes; (void)n_in; (void)out_size; (void)ws_size;
}